// GeneralIDConv_28793460752468
// MI455X (gfx1250) — compile-verified
//
#include <hip/hip_runtime.h>
#include <hip/hip_bf16.h>

typedef float v2f __attribute__((ext_vector_type(2)));
typedef float v8f __attribute__((ext_vector_type(8)));

#define C128 128

// ---------------- utility kernels ----------------

__global__ void copy_f4(const float4* __restrict__ src, float4* __restrict__ dst, int n4) {
    int i = blockIdx.x * blockDim.x + threadIdx.x;
    int stride = gridDim.x * blockDim.x;
    for (; i < n4; i += stride) dst[i] = src[i];
}

__global__ void zero_init(int* __restrict__ deg, int* __restrict__ meta, int n) {
    int i = blockIdx.x * blockDim.x + threadIdx.x;
    int stride = gridDim.x * blockDim.x;
    for (; i < n; i += stride) deg[i] = 0;
    if (blockIdx.x == 0 && threadIdx.x < 32) meta[threadIdx.x] = 0;
}

__global__ void label_hist(const int* __restrict__ lab, int* __restrict__ cnt, int n) {
    int i = blockIdx.x * blockDim.x + threadIdx.x;
    int stride = gridDim.x * blockDim.x;
    for (; i < n; i += stride) {
        int l = lab[i];
        if (l >= 0 && l < 8) atomicAdd(&cnt[l], 1);
    }
}

// meta layout: [0..7]=cnt, [8..15]=start, [16..23]=fill
__global__ void label_prefix(int* __restrict__ meta) {
    if (threadIdx.x == 0 && blockIdx.x == 0) {
        int run = 0;
        for (int l = 1; l < 8; ++l) {
            meta[8 + l]  = run;
            meta[16 + l] = run;
            run += meta[l];
        }
    }
}

__global__ void label_scatter(const int* __restrict__ lab, int* __restrict__ meta,
                              int* __restrict__ bucket, int n) {
    int i = blockIdx.x * blockDim.x + threadIdx.x;
    int stride = gridDim.x * blockDim.x;
    for (; i < n; i += stride) {
        int l = lab[i];
        if (l >= 1 && l < 8) {
            int p = atomicAdd(&meta[16 + l], 1);
            bucket[p] = i;
        }
    }
}

__global__ void deg_count(const int* __restrict__ row, int* __restrict__ deg, int e) {
    int i = blockIdx.x * blockDim.x + threadIdx.x;
    int stride = gridDim.x * blockDim.x;
    for (; i < e; i += stride) atomicAdd(&deg[row[i]], 1);
}

__global__ void make_dinv(const int* __restrict__ deg, float* __restrict__ dinv, int n) {
    int i = blockIdx.x * blockDim.x + threadIdx.x;
    int stride = gridDim.x * blockDim.x;
    for (; i < n; i += stride) {
        // +1 for the appended self loop; deg >= 1 always
        dinv[i] = rsqrtf((float)(deg[i] + 1));
    }
}

// out[v][j] = dinv[v]^2 * ox[v][j]   (self-loop term; also fully initializes d_out)
__global__ void selfloop_init(const float* __restrict__ ox, const float* __restrict__ dinv,
                              float* __restrict__ out, int n) {
    int i = blockIdx.x * blockDim.x + threadIdx.x;
    int stride = gridDim.x * blockDim.x;
    int total = n * C128;
    for (; i < total; i += stride) {
        int v = i >> 7;  // /128
        float d = dinv[v];
        out[i] = d * d * ox[i];
    }
}

// ---------------- WMMA gathered GEMM tile (16x16, K=128) ----------------
// Computes D = A_gathered x W for one 16x16 tile; caller provides row-index fetch.
// Layout assumptions (CDNA5 ISA 7.12.2, f32 16x16x4):
//   A: lane L holds row M=L&15, VGPR0/1 = K = k+(L>>4)*2, +1
//   B: lane L holds col N=L&15, VGPR0/1 = rows K = k+(L>>4)*2, +1
//   D: VGPR r -> M = r + (L>>4)*8, N = L&15
__device__ __forceinline__ v8f wmma_tile_f32(const float* __restrict__ arow,
                                             const float* __restrict__ wcol,
                                             int half) {
    v8f c = {};
#pragma unroll
    for (int k = 0; k < C128; k += 4) {
        int ka = k + half * 2;
        v2f a;
        a[0] = arow[ka];
        a[1] = arow[ka + 1];
        v2f b;
        b[0] = wcol[(size_t)ka * C128];
        b[1] = wcol[(size_t)(ka + 1) * C128];
        c = __builtin_amdgcn_wmma_f32_16x16x4_f32(false, a, false, b, (short)0, c,
                                                  false, false);
    }
    return c;
}

// Step 1: y = x[id] @ W ; x2[id[m]] += y[m]  (duplicate ids -> atomics)
__global__ void gemm_id_scatter(const float* __restrict__ x, const int* __restrict__ ids,
                                const float* __restrict__ W, float* __restrict__ x2,
                                int count) {
    int lane = threadIdx.x;           // blockDim.x == 32, one wave per block
    int half = lane >> 4;
    int lm   = lane & 15;
    int tile_m = blockIdx.x;
    int n0     = blockIdx.y * 16;

    int m = tile_m * 16 + lm;
    if (m >= count) m = count - 1;    // clamp (keep EXEC all-ones for WMMA)
    int gr = ids[m];

    v8f c = wmma_tile_f32(x + (size_t)gr * C128, W + n0 + lm, half);

    int mbase = tile_m * 16 + half * 8;
#pragma unroll
    for (int r = 0; r < 8; ++r) {
        int mm = mbase + r;
        if (mm < count) {
            int g = ids[mm];
            atomicAdd(&x2[(size_t)g * C128 + n0 + lm], c[r]);
        }
    }
}

// Step 2: for label l=z+1, rows=bucket[start[l]..]: ox[row] += x2[row] @ Wid[z]
__global__ void gemm_label(const float* __restrict__ x2, const int* __restrict__ bucket,
                           const int* __restrict__ meta, const float* __restrict__ Wid,
                           float* __restrict__ ox) {
    int l = blockIdx.z + 1;
    int count = meta[l];            // cnt[l]
    int start = meta[8 + l];        // start[l]
    int tile_m = blockIdx.x;
    if (tile_m * 16 >= count) return;   // wave-uniform exit

    int lane = threadIdx.x;
    int half = lane >> 4;
    int lm   = lane & 15;
    int n0   = blockIdx.y * 16;

    const int* rows = bucket + start;
    const float* W  = Wid + (size_t)blockIdx.z * C128 * C128;

    int m = tile_m * 16 + lm;
    if (m >= count) m = count - 1;
    int gr = rows[m];

    v8f c = wmma_tile_f32(x2 + (size_t)gr * C128, W + n0 + lm, half);

    int mbase = tile_m * 16 + half * 8;
#pragma unroll
    for (int r = 0; r < 8; ++r) {
        int mm = mbase + r;
        if (mm < count) {
            int g = rows[mm];
            ox[(size_t)g * C128 + n0 + lm] += c[r];   // rows unique per bucket: no atomic
        }
    }
}

// ---------------- edge scatter: out[col] += dinv[row]*dinv[col]*ox[row] ----------------
__global__ void edge_scatter(const int* __restrict__ row, const int* __restrict__ col,
                             const float* __restrict__ ox, const float* __restrict__ dinv,
                             float* __restrict__ out, int e) {
    int lane = threadIdx.x & 31;
    int wid  = blockIdx.x * (blockDim.x >> 5) + (threadIdx.x >> 5);
    int nw   = gridDim.x * (blockDim.x >> 5);
    for (int ed = wid; ed < e; ed += nw) {
        int r = row[ed];
        int c = col[ed];
        float w = dinv[r] * dinv[c];
        float4 v = ((const float4*)(ox + (size_t)r * C128))[lane];
        float* op = out + (size_t)c * C128 + lane * 4;
        atomicAdd(op + 0, w * v.x);
        atomicAdd(op + 1, w * v.y);
        atomicAdd(op + 2, w * v.z);
        atomicAdd(op + 3, w * v.w);
    }
}

// ---------------- launch ----------------
extern "C" void kernel_launch(void* const* d_in, const int* in_sizes, int n_in,
                              void* d_out, int out_size, void* d_ws, size_t ws_size,
                              hipStream_t stream) {
    const float* x   = (const float*)d_in[0];
    const int*   ei  = (const int*)d_in[1];
    const int*   ids = (const int*)d_in[2];
    const int*   lab = (const int*)d_in[3];
    const float* W   = (const float*)d_in[4];
    const float* Wid = (const float*)d_in[5];
    float* out = (float*)d_out;

    const int N   = in_sizes[3];
    const int E   = in_sizes[1] / 2;
    const int NID = in_sizes[2];
    const size_t NC = (size_t)N * C128;

    const int* row = ei;
    const int* col = ei + E;

    // workspace layout
    char* w = (char*)d_ws;
    float* x2   = (float*)w;                 w += NC * sizeof(float);
    float* ox   = (float*)w;                 w += NC * sizeof(float);
    float* dinv = (float*)w;                 w += (size_t)N * sizeof(float);
    int*   deg  = (int*)w;                   w += (size_t)N * sizeof(int);
    int*   meta = (int*)w;                   w += 32 * sizeof(int);   // cnt[8], start[8], fill[8]
    int*   bucket = (int*)w;                 w += (size_t)N * sizeof(int);
    (void)ws_size; (void)n_in; (void)out_size;

    const int n4 = (int)(NC / 4);

    // 1) x2 = x
    copy_f4<<<dim3(1024), dim3(256), 0, stream>>>((const float4*)x, (float4*)x2, n4);
    // 2) zero deg + meta
    zero_init<<<dim3((N + 255) / 256), dim3(256), 0, stream>>>(deg, meta, N);
    // 3) x2[id] += x[id] @ W   (WMMA, atomic scatter)
    gemm_id_scatter<<<dim3((NID + 15) / 16, C128 / 16), dim3(32), 0, stream>>>(
        x, ids, W, x2, NID);
    // 4) ox = x2
    copy_f4<<<dim3(1024), dim3(256), 0, stream>>>((const float4*)x2, (float4*)ox, n4);
    // 5-7) counting sort of nodes by label
    label_hist<<<dim3((N + 255) / 256), dim3(256), 0, stream>>>(lab, meta, N);
    label_prefix<<<dim3(1), dim3(32), 0, stream>>>(meta);
    label_scatter<<<dim3((N + 255) / 256), dim3(256), 0, stream>>>(lab, meta, bucket, N);
    // 8-9) degrees (row occurrences + self loop) -> dinv
    deg_count<<<dim3((E + 255) / 256), dim3(256), 0, stream>>>(row, deg, E);
    make_dinv<<<dim3((N + 255) / 256), dim3(256), 0, stream>>>(deg, dinv, N);
    // 10) grouped GEMM per label: ox[rows] += x2[rows] @ Wid[l]   (WMMA)
    gemm_label<<<dim3((N + 15) / 16, C128 / 16, 7), dim3(32), 0, stream>>>(
        x2, bucket, meta, Wid, ox);
    // 11) out = dinv^2 * ox   (self loops; fully writes d_out)
    selfloop_init<<<dim3((int)((NC + 255) / 256)), dim3(256), 0, stream>>>(ox, dinv, out, N);
    // 12) out[col] += dinv[row]*dinv[col]*ox[row]
    edge_scatter<<<dim3((E + 7) / 8), dim3(256), 0, stream>>>(row, col, ox, dinv, out, E);
}